// VEMAU_38817914422019
// MI455X (gfx1250) — compile-verified
//
#include <hip/hip_runtime.h>

// ---------------------------------------------------------------------------
// EMA unit for MI455X (gfx1250, wave32). All GEMMs on V_WMMA_F32_16X16X4_F32.
// Conv GEMMs use double-buffered ASYNC global->LDS staging (ASYNCcnt).
//
// f32 WMMA fragment layout (per CDNA5 ISA 7.12.2):
//   A (16x4, MxK):  lane = {m = lane&15, half = lane>>4};
//                   a.x = A[m][2*half], a.y = A[m][2*half+1]
//   B (4x16, KxN):  b.x = B[2*half][n=lane&15], b.y = B[2*half+1][n]
//   C/D (16x16):    reg r, lane -> D[r + 8*half][lane&15]
// ---------------------------------------------------------------------------

typedef float v2f __attribute__((ext_vector_type(2)));
typedef float v8f __attribute__((ext_vector_type(8)));

#define WMMA_F32(acc, a, b) \
  (acc) = __builtin_amdgcn_wmma_f32_16x16x4_f32(false, (a), false, (b), (short)0, (acc), false, false)

static __device__ __forceinline__ v8f vzero8() {
  v8f z = {0.f, 0.f, 0.f, 0.f, 0.f, 0.f, 0.f, 0.f};
  return z;
}

constexpr int B_ = 16;    // batch
constexpr int C_ = 512;   // channels
constexpr int K_ = 64;    // EM bases
constexpr int N_ = 4096;  // H*W
constexpr int NCHUNK = 8; // split-K chunks for mu-update GEMM

// --------------------------- async global -> LDS ---------------------------
#if __has_builtin(__builtin_amdgcn_global_load_async_to_lds_b128) && \
    __has_builtin(__builtin_amdgcn_global_load_async_to_lds_b32)
#define HAVE_ASYNC_LDS 1
#else
#define HAVE_ASYNC_LDS 0
#endif

// Parameter types per hipcc diagnostics (HIP prints AS1 as "__device__"):
//   b32 : (int AS1*, int AS3*, Ii offset, Ii cpol)
//   b128: (int __vector(16B) AS1*, ... AS3*, Ii, Ii)
typedef int b128i __attribute__((vector_size(16)));
typedef __attribute__((address_space(1))) int*   as1_i32p;
typedef __attribute__((address_space(3))) int*   as3_i32p;
typedef __attribute__((address_space(1))) b128i* as1_b128p;
typedef __attribute__((address_space(3))) b128i* as3_b128p;

static __device__ __forceinline__ void cp_b32(const float* g, float* l) {
#if HAVE_ASYNC_LDS
  __builtin_amdgcn_global_load_async_to_lds_b32((as1_i32p)g, (as3_i32p)l, 0, 0);
#else
  *l = *g;
#endif
}

static __device__ __forceinline__ void cp_b128(const float* g, float* l) {
#if HAVE_ASYNC_LDS
  __builtin_amdgcn_global_load_async_to_lds_b128((as1_b128p)g, (as3_b128p)l, 0, 0);
#else
  *(float4*)l = *(const float4*)g;
#endif
}

static __device__ __forceinline__ void wait_async0() {
#if HAVE_ASYNC_LDS
#if __has_builtin(__builtin_amdgcn_s_wait_asynccnt)
  __builtin_amdgcn_s_wait_asynccnt(0);
#else
  asm volatile("s_wait_asynccnt 0x0" ::: "memory");
#endif
#endif
}

// ---------------------------------------------------------------------------
// 1x1 conv as GEMM: Y[b][o][n] = sum_c W[o][c] * X[b][c][n] (+ bias[o])
// Block: 256 thr / 8 waves; tile M=128(o) x N=64(n); K staged 16-wide in LDS,
// double-buffered with async global->LDS copies.
//
// A tile LDS layout: K-pair interleaved [k>>1][m][k&1], pitch 288 dwords
//   (row delta of the two 16-lane halves lands 32 banks apart; fragment load
//    is a single ds_load_b64).
// B tile LDS layout: row-contiguous [k][n], pitch 80 dwords (rows stay 16B
//   aligned and memory-contiguous so the b128 async copy lands directly).
// ---------------------------------------------------------------------------
__global__ __launch_bounds__(256)
void k_conv(const float* __restrict__ X, const float* __restrict__ W,
            const float* __restrict__ bias, float* __restrict__ Y, int hasBias) {
  constexpr int LDA = 288;            // dwords per A k-pair row (128*2 + 32 pad)
  constexpr int LDB = 80;             // dwords per B k row (64 + 16 pad)
  constexpr int NSTAGE = C_ / 16;     // 32 K-stages
  __shared__ float lA[2][8 * LDA];
  __shared__ float lB[2][16 * LDB];

  const int b    = blockIdx.z;
  const int o0   = blockIdx.y * 128;
  const int n0   = blockIdx.x * 64;
  const int t    = threadIdx.x;
  const int wv   = t >> 5;
  const int lane = t & 31;
  const int m    = lane & 15;
  const int half = lane >> 4;
  const float* Xb = X + (size_t)b * C_ * N_;

  v8f acc[4];
  for (int j = 0; j < 4; ++j) acc[j] = vzero8();

  auto stage = [&](int c0, int buf) {
    float* dA = &lA[buf][0];
    float* dB = &lB[buf][0];
    // A: 128(o) x 16(c) elements, 8 per thread, scattered into pair layout
    for (int i = 0; i < 8; ++i) {
      const int e  = t + 256 * i;
      const int kk = e & 15;
      const int mm = e >> 4;
      cp_b32(&W[(size_t)(o0 + mm) * C_ + c0 + kk],
             &dA[(kk >> 1) * LDA + mm * 2 + (kk & 1)]);
    }
    // B: 16(c) x 64(n), one 16B chunk per thread, memory-layout preserved
    {
      const int kk = t >> 4;
      const int nn = (t & 15) * 4;
      cp_b128(&Xb[(size_t)(c0 + kk) * N_ + n0 + nn], &dB[kk * LDB + nn]);
    }
  };

  stage(0, 0);
  for (int s = 0; s < NSTAGE; ++s) {
    wait_async0();        // my stage-s tiles are in LDS
    __syncthreads();      // everyone's stage-s tiles visible; stage s-1 reads done
    if (s + 1 < NSTAGE) stage((s + 1) * 16, (s + 1) & 1);

    const float* cA = &lA[s & 1][0];
    const float* cB = &lB[s & 1][0];
    for (int k4 = 0; k4 < 16; k4 += 4) {
      const v2f a = *(const v2f*)&cA[(k4 / 2 + half) * LDA + (16 * wv + m) * 2];
      for (int j = 0; j < 4; ++j) {
        v2f bf;
        bf.x = cB[(k4 + 2 * half + 0) * LDB + 16 * j + m];
        bf.y = cB[(k4 + 2 * half + 1) * LDB + 16 * j + m];
        WMMA_F32(acc[j], a, bf);
      }
    }
  }

  float* Yb = Y + (size_t)b * C_ * N_;
  for (int j = 0; j < 4; ++j) {
    for (int r = 0; r < 8; ++r) {
      const int row = o0 + 16 * wv + r + 8 * half;
      const int col = n0 + 16 * j + m;
      float v = acc[j][r];
      if (hasBias) v += bias[row];
      Yb[(size_t)row * N_ + col] = v;
    }
  }
}

// ---------------------------------------------------------------------------
// z = X^T * mu, then in-register V-E step:
//   z = 0.5*softmax_k(z) + 0.5 * 64 / (63 + 2*||z||_k)
// Wave owns 16 n-rows x full 64 k -> row reductions via half-wave shuffles.
// Grid: (N_/128, B_)
// ---------------------------------------------------------------------------
__global__ __launch_bounds__(256)
void k_zve(const float* __restrict__ XS, const float* __restrict__ MU,
           int muBatchStride, float* __restrict__ Z) {
  const int b    = blockIdx.y;
  const int t    = threadIdx.x;
  const int wv   = t >> 5;
  const int lane = t & 31;
  const int m    = lane & 15;
  const int half = lane >> 4;
  const int n0   = blockIdx.x * 128 + wv * 16;

  const float* Xb = XS + (size_t)b * C_ * N_;
  const float* Mb = MU + (size_t)b * muBatchStride;

  v8f acc[4];
  for (int j = 0; j < 4; ++j) acc[j] = vzero8();

  for (int c = 0; c < C_; c += 4) {
    v2f a;  // A[m=n][kk=c]: lanes stride-1 in n (coalesced)
    a.x = Xb[(size_t)(c + 2 * half + 0) * N_ + n0 + m];
    a.y = Xb[(size_t)(c + 2 * half + 1) * N_ + n0 + m];
    for (int j = 0; j < 4; ++j) {
      v2f bf;  // B[kk=c][col=k]: mu is tiny, L2-resident
      bf.x = Mb[(c + 2 * half + 0) * K_ + 16 * j + m];
      bf.y = Mb[(c + 2 * half + 1) * K_ + 16 * j + m];
      WMMA_F32(acc[j], a, bf);
    }
  }

  // V-E epilogue: row n = n0 + r + 8*half, cols k = 16*j + m
  const float alpha0 = 0.5f;  // sin^2(45 deg)
  const float beta0  = 0.5f;  // cos^2(45 deg)
  for (int r = 0; r < 8; ++r) {
    float mx = -3.0e38f, ssq = 0.f;
    for (int j = 0; j < 4; ++j) {
      float v = acc[j][r];
      mx  = fmaxf(mx, v);
      ssq += v * v;
    }
    for (int s = 1; s < 16; s <<= 1) {
      mx  = fmaxf(mx, __shfl_xor(mx, s));
      ssq += __shfl_xor(ssq, s);
    }
    float e[4];
    float se = 0.f;
    for (int j = 0; j < 4; ++j) {
      e[j] = __expf(acc[j][r] - mx);
      se += e[j];
    }
    for (int s = 1; s < 16; s <<= 1) se += __shfl_xor(se, s);

    const float add  = beta0 * (float)K_ / ((float)(K_ - 1) + 2.f * sqrtf(ssq));
    const float sc   = alpha0 / se;
    const int   nrow = n0 + r + 8 * half;
    for (int j = 0; j < 4; ++j) {
      Z[((size_t)b * N_ + nrow) * K_ + 16 * j + m] = e[j] * sc + add;
    }
  }
}

// ---------------------------------------------------------------------------
// Deterministic per-(b,k) column reduction: R[b][k] = sum_r f(S[b][r][k])
// sq=0 -> identity (z-sums over n); sq=1 -> square (mu sum-of-squares over c)
// ---------------------------------------------------------------------------
__global__ __launch_bounds__(256)
void k_colreduce(const float* __restrict__ S, float* __restrict__ R, int rows, int sq) {
  const int b = blockIdx.x;
  const float* Sb = S + (size_t)b * rows * K_;
  __shared__ float part[256];
  const int k = threadIdx.x & 63;
  const int g = threadIdx.x >> 6;
  float a = 0.f;
  for (int r = g; r < rows; r += 4) {
    float v = Sb[(size_t)r * K_ + k];
    a += sq ? v * v : v;
  }
  part[threadIdx.x] = a;
  __syncthreads();
  if (g == 0) R[b * K_ + k] = part[k] + part[64 + k] + part[128 + k] + part[192 + k];
}

// ---------------------------------------------------------------------------
// mu-update GEMM (split-K over n): MUpart[chunk][b][c][k] = sum_{n in chunk} xs*z
// Grid: (C_/128, NCHUNK, B_). A-stream is per-lane strided (stride N_), so
// prefetch ahead on each lane's private row.
// ---------------------------------------------------------------------------
__global__ __launch_bounds__(256)
void k_mu(const float* __restrict__ XS, const float* __restrict__ Z,
          float* __restrict__ MUpart) {
  const int b     = blockIdx.z;
  const int chunk = blockIdx.y;
  const int t     = threadIdx.x;
  const int wv    = t >> 5;
  const int lane  = t & 31;
  const int m     = lane & 15;
  const int half  = lane >> 4;
  const int c0    = blockIdx.x * 128 + wv * 16;

  const float* Xb = XS + (size_t)b * C_ * N_;
  const float* Zb = Z + (size_t)b * N_ * K_;

  v8f acc[4];
  for (int j = 0; j < 4; ++j) acc[j] = vzero8();

  const int nBeg = chunk * (N_ / NCHUNK);
  const int nEnd = nBeg + (N_ / NCHUNK);
  const float* aRow = &Xb[(size_t)(c0 + m) * N_];
  for (int n = nBeg; n < nEnd; n += 4) {
    if ((n & 63) == 0)  // one cacheline-granular prefetch per lane-row per 64 n
      __builtin_prefetch(aRow + n + 128, 0, 1);
    // A[m=c][kk=n]: consecutive along n -> one 8B load per lane
    v2f a = *(const v2f*)&aRow[n + 2 * half];
    for (int j = 0; j < 4; ++j) {
      v2f bf;  // B[kk=n][col=k]: lanes contiguous in k
      bf.x = Zb[(size_t)(n + 2 * half + 0) * K_ + 16 * j + m];
      bf.y = Zb[(size_t)(n + 2 * half + 1) * K_ + 16 * j + m];
      WMMA_F32(acc[j], a, bf);
    }
  }

  float* P = MUpart + (size_t)(chunk * B_ + b) * C_ * K_;
  for (int j = 0; j < 4; ++j)
    for (int r = 0; r < 8; ++r)
      P[(size_t)(c0 + r + 8 * half) * K_ + 16 * j + m] = acc[j][r];
}

// Combine split-K partials; apply z_ = z / (1e-6 + sum_n z) scaling.
__global__ void k_mufinal(const float* __restrict__ MUpart, const float* __restrict__ zsum,
                          float* __restrict__ MUraw) {
  const int i = blockIdx.x * 256 + threadIdx.x;
  if (i >= B_ * C_ * K_) return;
  const int k  = i & (K_ - 1);
  const int bb = i / (C_ * K_);
  float s = 0.f;
  for (int c = 0; c < NCHUNK; ++c) s += MUpart[(size_t)c * B_ * C_ * K_ + i];
  MUraw[i] = s / (1e-6f + zsum[bb * K_ + k]);
}

// mu = mu_raw / (1e-6 + ||mu_raw||_c); optionally emit the mu output tensor.
__global__ void k_munorm(const float* __restrict__ MUraw, const float* __restrict__ musq,
                         float* __restrict__ MU, float* __restrict__ outMu, int writeOut) {
  const int i = blockIdx.x * 256 + threadIdx.x;
  if (i >= B_ * C_ * K_) return;
  const int k  = i & (K_ - 1);
  const int bb = i / (C_ * K_);
  const float s = 1.0f / (1e-6f + sqrtf(musq[bb * K_ + k]));
  const float v = MUraw[i] * s;
  MU[i] = v;
  if (writeOut) outMu[i] = v;
}

// ---------------------------------------------------------------------------
// Reconstruction: xr[b][c][n] = relu(sum_k mu[b][c][k] * z[b][n][k])
// z tile (16 n x 64 k) transposed through LDS -> clean ds_load B-fragments.
// Grid: (N_/16, C_/128, B_)
// ---------------------------------------------------------------------------
__global__ __launch_bounds__(256)
void k_recon(const float* __restrict__ MU, const float* __restrict__ Z,
             float* __restrict__ XR) {
  __shared__ float lB[64 * 16];  // [k][n]
  const int b    = blockIdx.z;
  const int c0   = blockIdx.y * 128;
  const int n0   = blockIdx.x * 16;
  const int t    = threadIdx.x;
  const int wv   = t >> 5;
  const int lane = t & 31;
  const int m    = lane & 15;
  const int half = lane >> 4;

  const float* Zb = Z + (size_t)b * N_ * K_;
  const float* Mb = MU + (size_t)b * C_ * K_;

  {  // stage z tile: coalesced float4 reads, transposed LDS writes
    const int nn = t >> 4;
    const int kk = (t & 15) * 4;
    const float4 v = *(const float4*)&Zb[(size_t)(n0 + nn) * K_ + kk];
    lB[(kk + 0) * 16 + nn] = v.x;
    lB[(kk + 1) * 16 + nn] = v.y;
    lB[(kk + 2) * 16 + nn] = v.z;
    lB[(kk + 3) * 16 + nn] = v.w;
  }
  __syncthreads();

  v8f acc = vzero8();
  const int crow = c0 + wv * 16 + m;
  for (int k4 = 0; k4 < K_; k4 += 4) {
    v2f a = *(const v2f*)&Mb[(size_t)crow * K_ + k4 + 2 * half];
    v2f bf;
    bf.x = lB[(k4 + 2 * half + 0) * 16 + m];
    bf.y = lB[(k4 + 2 * half + 1) * 16 + m];
    WMMA_F32(acc, a, bf);
  }

  float* Xb = XR + (size_t)b * C_ * N_;
  for (int r = 0; r < 8; ++r) {
    const int row = c0 + wv * 16 + r + 8 * half;
    Xb[(size_t)row * N_ + n0 + m] = fmaxf(acc[r], 0.0f);
  }
}

// ---------------------------------------------------------------------------
// BN training-mode stats: one block per channel (deterministic).
// ---------------------------------------------------------------------------
__global__ __launch_bounds__(256)
void k_bnstat(const float* __restrict__ Y, float* __restrict__ meanv,
              float* __restrict__ varv) {
  const int o = blockIdx.x;
  __shared__ float s1[256];
  __shared__ float s2[256];
  float a = 0.f, q = 0.f;
  for (int bb = 0; bb < B_; ++bb) {
    const float* p = Y + ((size_t)bb * C_ + o) * N_;
    for (int n = threadIdx.x; n < N_; n += 256) {
      float v = p[n];
      a += v;
      q += v * v;
    }
  }
  s1[threadIdx.x] = a;
  s2[threadIdx.x] = q;
  __syncthreads();
  for (int s = 128; s > 0; s >>= 1) {
    if (threadIdx.x < s) {
      s1[threadIdx.x] += s1[threadIdx.x + s];
      s2[threadIdx.x] += s2[threadIdx.x + s];
    }
    __syncthreads();
  }
  if (threadIdx.x == 0) {
    const float cnt = (float)(B_ * N_);
    const float mu  = s1[0] / cnt;
    meanv[o] = mu;
    varv[o]  = s2[0] / cnt - mu * mu;
  }
}

// BN apply + residual + ReLU, in-place on d_out; also writes theta scalar.
__global__ void k_bnapply(const float* __restrict__ X, const float* __restrict__ gamma,
                          const float* __restrict__ beta, const float* __restrict__ meanv,
                          const float* __restrict__ varv, float* __restrict__ OUT,
                          float* __restrict__ theta) {
  const size_t i = (size_t)blockIdx.x * 256 + threadIdx.x;
  if (i == 0) *theta = 45.0f;
  if (i >= (size_t)B_ * C_ * N_) return;
  const int o = (int)((i / N_) % C_);
  const float y = OUT[i];
  const float v = gamma[o] * (y - meanv[o]) * rsqrtf(varv[o] + 1e-5f) + beta[o] + X[i];
  OUT[i] = fmaxf(v, 0.f);
}

// ---------------------------------------------------------------------------
extern "C" void kernel_launch(void* const* d_in, const int* in_sizes, int n_in,
                              void* d_out, int out_size, void* d_ws, size_t ws_size,
                              hipStream_t stream) {
  const float* x     = (const float*)d_in[0];  // [16,512,64,64]
  const float* W1    = (const float*)d_in[1];  // [512,512] (o,c)
  const float* b1    = (const float*)d_in[2];  // [512]
  const float* W2    = (const float*)d_in[3];  // [512,512]
  const float* gamma = (const float*)d_in[4];  // [512]
  const float* beta  = (const float*)d_in[5];  // [512]
  const float* mu0   = (const float*)d_in[6];  // [1,512,64]

  float* out   = (float*)d_out;                 // [16,512,4096]
  float* outMu = out + (size_t)B_ * C_ * N_;    // [16,512,64]
  float* outTh = outMu + (size_t)B_ * C_ * K_;  // [1]

  float* ws     = (float*)d_ws;
  float* xc     = ws;                                     // 33,554,432 (xs, later xr)
  float* z      = xc + (size_t)B_ * C_ * N_;              //  4,194,304
  float* mupart = z + (size_t)B_ * N_ * K_;               //  4,194,304
  float* muraw  = mupart + (size_t)NCHUNK * B_ * C_ * K_; //    524,288
  float* mubuf  = muraw + (size_t)B_ * C_ * K_;           //    524,288
  float* zsum   = mubuf + (size_t)B_ * C_ * K_;           //      1,024
  float* musq   = zsum + B_ * K_;                         //      1,024
  float* meanv  = musq + B_ * K_;                         //        512
  float* varv   = meanv + C_;                             //        512

  // conv1: xc = W1 * x + b1
  k_conv<<<dim3(N_ / 64, C_ / 128, B_), 256, 0, stream>>>(x, W1, b1, xc, 1);

  // EM loop (2 stages)
  for (int s = 0; s < 2; ++s) {
    const float* musrc = (s == 0) ? mu0 : mubuf;
    const int mstride  = (s == 0) ? 0 : C_ * K_;
    k_zve<<<dim3(N_ / 128, B_), 256, 0, stream>>>(xc, musrc, mstride, z);
    k_colreduce<<<B_, 256, 0, stream>>>(z, zsum, N_, 0);
    k_mu<<<dim3(C_ / 128, NCHUNK, B_), 256, 0, stream>>>(xc, z, mupart);
    k_mufinal<<<(B_ * C_ * K_ + 255) / 256, 256, 0, stream>>>(mupart, zsum, muraw);
    k_colreduce<<<B_, 256, 0, stream>>>(muraw, musq, C_, 1);
    k_munorm<<<(B_ * C_ * K_ + 255) / 256, 256, 0, stream>>>(muraw, musq, mubuf, outMu,
                                                             s == 1);
  }

  // reconstruction (xs is dead; reuse its buffer for xr)
  k_recon<<<dim3(N_ / 16, C_ / 128, B_), 256, 0, stream>>>(mubuf, z, xc);

  // conv2: out = W2 * xr (BN applied afterwards)
  k_conv<<<dim3(N_ / 64, C_ / 128, B_), 256, 0, stream>>>(xc, W2, nullptr, out, 0);

  // BN stats + apply (+ residual + ReLU + theta)
  k_bnstat<<<C_, 256, 0, stream>>>(out, meanv, varv);
  const size_t tot = (size_t)B_ * C_ * N_;
  k_bnapply<<<(unsigned)((tot + 255) / 256), 256, 0, stream>>>(x, gamma, beta, meanv, varv,
                                                               out, outTh);
}